// GroupedLinear_44367012168433
// MI455X (gfx1250) — compile-verified
//
#include <hip/hip_runtime.h>

// ---------------------------------------------------------------------------
// GroupedLinear (per-group LayerNorm + per-group 64x64 Linear, transposed out)
//
// x   : [1024, 8, 64, 64] f32  -> per group g: 32768 rows x 64 contiguous f32
// out : [1024, 8, 64, 64] f32  -> out[(g*64+o)*32768 + r] = y_r[o]
//
// f32 WMMA path: D(16o x 16r) = W_tile(16o x K) * Xn^T(K x 16r) + bias,
// chained V_WMMA_F32_16X16X4_F32 over K=64 (16 slices of 4).
// ---------------------------------------------------------------------------

typedef float v2f __attribute__((ext_vector_type(2)));
typedef float v4f __attribute__((ext_vector_type(4)));
typedef float v8f __attribute__((ext_vector_type(8)));

#define NGROUP   16
#define ING      64
#define OUTG     64
#define RPG      32768          // rows per group = 8*64*64
#define CHROWS   64             // rows normalized per block iteration
#define NCHUNK   4              // chunks per block
#define LSTRIDE  68             // LDS row stride (floats): 16B-aligned rows,
                                // conflict-free b64 B-reads (bank = 4*lo+off)

__launch_bounds__(128, 1)
__global__ void grouped_ln_linear_wmma(const float* __restrict__ x,
                                       const float* __restrict__ gamma,
                                       const float* __restrict__ beta,
                                       const float* __restrict__ W,
                                       const float* __restrict__ bias,
                                       float* __restrict__ out) {
    __shared__ float sXn[CHROWS * LSTRIDE];   // normalized tile [64 rows][68]
    __shared__ float sG[ING];
    __shared__ float sB[ING];

    const int t     = threadIdx.x;            // 0..127, 4 waves (wave32)
    const int bid   = blockIdx.x;             // 2048 blocks: 128 per group
    const int g     = bid >> 7;
    const int cbase = (bid & 127) * NCHUNK;

    // Stage LN params in LDS (broadcast reads later).
    if (t < ING) sG[t] = gamma[g * ING + t];
    else         sB[t - ING] = beta[g * ING + (t - ING)];

    // ---- phase-2 identities ----
    const int w  = t >> 5;                    // wave id == o-tile (0..3)
    const int l  = t & 31;
    const int lo = l & 15;
    const int hi = l >> 4;

    // A matrix (ISA 16x4 f32 layout): lane = M = o within tile,
    // VGPR0/1 hold K = 4*kk + 2*hi + {0,1}. Contiguous float2 from W row.
    v2f A[16];
    {
        const float* wp = W + ((size_t)g * OUTG + (size_t)(w * 16 + lo)) * ING
                            + 2 * hi;
        #pragma unroll
        for (int kk = 0; kk < 16; ++kk)
            A[kk] = *(const v2f*)(wp + 4 * kk);
    }

    // Bias in C layout: VGPR v holds o = w*16 + hi*8 + v (same for all lanes
    // in the half-wave -> broadcast loads).
    v8f cinit;
    #pragma unroll
    for (int v = 0; v < 8; ++v)
        cinit[v] = bias[g * OUTG + w * 16 + hi * 8 + v];

    // ---- phase-1 identities ----
    const int row  = t >> 1;                  // 0..63
    const int half = t & 1;                   // which 32 of the 64 elems
    const float* gbase = x + (size_t)g * ((size_t)RPG * ING);

    __syncthreads();                          // sG/sB visible

    for (int c = 0; c < NCHUNK; ++c) {
        const int chunk = cbase + c;

        // ================= Phase 1: LayerNorm 64 rows =================
        const float* xr = gbase + (size_t)chunk * (CHROWS * ING)
                        + row * ING + half * 32;
        v4f vv[8];
        float s = 0.f, ss = 0.f;
        #pragma unroll
        for (int i = 0; i < 8; ++i) {
            vv[i] = *(const v4f*)(xr + 4 * i);      // 8 x b128, one 128B line
            #pragma unroll
            for (int j = 0; j < 4; ++j) {
                float e = vv[i][j];
                s  += e;
                ss += e * e;
            }
        }
        // partner lane (t^1) holds the other half of this row
        s  += __shfl_xor(s, 1, 32);
        ss += __shfl_xor(ss, 1, 32);
        const float mean = s * (1.0f / 64.0f);
        const float var  = ss * (1.0f / 64.0f) - mean * mean;
        const float rstd = rsqrtf(var + 1e-6f);

        float*       dst = &sXn[row * LSTRIDE + half * 32];
        const float* gg  = &sG[half * 32];
        const float* bb  = &sB[half * 32];
        #pragma unroll
        for (int i = 0; i < 8; ++i) {
            v4f gv = *(const v4f*)(gg + 4 * i);
            v4f bv = *(const v4f*)(bb + 4 * i);
            v4f o  = (vv[i] - mean) * rstd * gv + bv;
            *(v4f*)(dst + 4 * i) = o;               // aligned ds_store_b128
        }

        __syncthreads();

        // ================= Phase 2: WMMA GEMM =================
        // out[(g*64 + o)*RPG + chunk*64 + rt*16 + lo], o = w*16 + hi*8 + v
        float* outg = out + ((size_t)g * OUTG + (size_t)(w * 16 + hi * 8)) * RPG
                    + (size_t)chunk * CHROWS + lo;

        #pragma unroll
        for (int rt = 0; rt < 4; ++rt) {
            v8f acc = cinit;
            // B (4x16 f32): lane = N = r, VGPR0/1 = K = 4*kk + 2*hi + {0,1}
            const float* bp = &sXn[(rt * 16 + lo) * LSTRIDE + 2 * hi];
            #pragma unroll
            for (int kk = 0; kk < 16; ++kk) {
                v2f bv = *(const v2f*)(bp + 4 * kk); // conflict-free ds_load_b64
                acc = __builtin_amdgcn_wmma_f32_16x16x4_f32(
                        false, A[kk], false, bv, (short)0, acc, false, false);
            }
            #pragma unroll
            for (int v = 0; v < 8; ++v)
                outg[(size_t)v * RPG + rt * 16] = acc[v];  // 64B/half-wave,
                                                           // L2-merged to 128B
        }

        __syncthreads();
    }
}

extern "C" void kernel_launch(void* const* d_in, const int* in_sizes, int n_in,
                              void* d_out, int out_size, void* d_ws, size_t ws_size,
                              hipStream_t stream) {
    const float* x  = (const float*)d_in[0];
    const float* g  = (const float*)d_in[1];
    const float* be = (const float*)d_in[2];
    const float* W  = (const float*)d_in[3];
    const float* b  = (const float*)d_in[4];
    float* out = (float*)d_out;

    dim3 grid(NGROUP * (RPG / CHROWS / NCHUNK));  // 16 * 128 = 2048 blocks
    dim3 block(128);                              // 4 waves (wave32)
    hipLaunchKernelGGL(grouped_ln_linear_wmma, grid, block, 0, stream,
                       x, g, be, W, b, out);
}